// SKVQuantProcessor_82978768159600
// MI455X (gfx1250) — compile-verified
//
#include <hip/hip_runtime.h>
#include <hip/hip_bf16.h>

// Problem constants (match reference setup_inputs)
#define BS      4
#define NH      32
#define SL      4096
#define HD      128
#define HIDDEN  (NH * HD)          // 4096
#define GSIZE   128
#define NGROUPS (HIDDEN / GSIZE)   // 32
#define TOKENS  (BS * SL)          // 16384
#define CLIPF   0.96f
#define EPSF    1e-5f
#define MAXQF   3.0f

#define NTHREADS 256
#define NBLOCKS  1024

// GCC-vector int4: the pointee type the gfx1250 async builtins expect.
typedef int v4i __attribute__((vector_size(16)));
typedef __attribute__((address_space(1))) v4i gv4i;   // global
typedef __attribute__((address_space(3))) v4i lv4i;   // LDS

// ---- CDNA5 async global<->LDS helpers (ASYNCcnt-tracked) ----------------

__device__ __forceinline__ void async_load16(const float* gsrc, float* ldst) {
#if __has_builtin(__builtin_amdgcn_global_load_async_to_lds_b128)
  __builtin_amdgcn_global_load_async_to_lds_b128((gv4i*)gsrc, (lv4i*)ldst, 0, 0);
#else
  unsigned loff = (unsigned)(unsigned long long)ldst; // low 32 bits = LDS offset
  asm volatile("global_load_async_to_lds_b128 %0, %1, off"
               :: "v"(loff), "v"(gsrc) : "memory");
#endif
}

__device__ __forceinline__ void async_store16(float* gdst, const float* lsrc) {
#if __has_builtin(__builtin_amdgcn_global_store_async_from_lds_b128)
  __builtin_amdgcn_global_store_async_from_lds_b128((gv4i*)gdst, (lv4i*)lsrc, 0, 0);
#else
  unsigned loff = (unsigned)(unsigned long long)lsrc;
  asm volatile("global_store_async_from_lds_b128 %0, %1, off"
               :: "v"(gdst), "v"(loff) : "memory");
#endif
}

__device__ __forceinline__ void wait_async0() {
#if __has_builtin(__builtin_amdgcn_s_wait_asynccnt)
  __builtin_amdgcn_s_wait_asynccnt(0);
#else
  asm volatile("s_wait_asynccnt 0" ::: "memory");
#endif
}

// ---- Kernel -------------------------------------------------------------
//
// One token (b, s) has hidden vector x[c] stored at
//   tensor[b, c>>7, s, c&127]  (c = original channel 0..4095)
// Per token:
//   v[p]   = x[ridx[p]] * ss[ridx[p]]        (p = reordered channel)
//   group quant/dequant over 128 consecutive p
//   out[ridx[p]] = deq[p] / ss[ridx[p]]      (permutation => direct scatter)
//
// Wave32 mapping: group g handled by one wave, 4 channels per lane.

__global__ void __launch_bounds__(NTHREADS)
skvq_kernel(const float* __restrict__ x,
            const float* __restrict__ ss,
            const int*   __restrict__ ridx,
            float* __restrict__ out) {
  __shared__ float xbuf[HIDDEN];    // 16 KB token staging (in-place dequant)
  __shared__ int   ridx_s[HIDDEN];  // 16 KB permutation
  __shared__ float mult_s[HIDDEN];  // 16 KB  ss[ridx[p]]

  const int tid = threadIdx.x;

  // One-time per block: permutation + gathered smooth scale (tiny, L2-hot)
  for (int c = tid; c < HIDDEN; c += NTHREADS) {
    int r = ridx[c];
    ridx_s[c] = r;
    mult_s[c] = ss[r];
  }
  __syncthreads();

  const int wave = tid >> 5;
  const int lane = tid & 31;

  for (int t = blockIdx.x; t < TOKENS; t += gridDim.x) {
    const int b = t >> 12;         // t / SL
    const int s = t & (SL - 1);    // t % SL
    const size_t base = (size_t)b * ((size_t)NH * SL * HD) + (size_t)s * HD;

    // ---- Stage 1: async-copy token into LDS (coalesced 512B segments) ----
    // 1024 float4 chunks; wave w iteration k covers one contiguous head row.
#pragma unroll
    for (int k = 0; k < 4; ++k) {
      int v    = tid + k * NTHREADS;      // float4 index 0..1023
      int nh_i = v >> 5;                  // head index (chunk of 128 floats)
      int off  = (v & 31) * 4;            // offset within head row
      const float* gp = x + base + (size_t)nh_i * ((size_t)SL * HD) + off;
      async_load16(gp, &xbuf[v * 4]);
    }
    wait_async0();
    __syncthreads();

    // ---- Stage 2: per-group 2-bit quant/dequant, scatter in place --------
#pragma unroll
    for (int i = 0; i < 4; ++i) {
      const int g     = wave + i * 8;         // group id 0..31
      const int cbase = g * GSIZE + lane;     // reordered channel base

      float val[4], mm[4];
      int   rr[4];
      float vmin =  3.402823466e38f;
      float vmax = -3.402823466e38f;
#pragma unroll
      for (int j = 0; j < 4; ++j) {
        int   c = cbase + j * 32;
        int   r = ridx_s[c];
        float m = mult_s[c];
        float v = xbuf[r] * m;               // smooth + gather
        rr[j] = r; mm[j] = m; val[j] = v;
        vmin = fminf(vmin, v);
        vmax = fmaxf(vmax, v);
      }
      // wave32 cross-lane min/max reduction
#pragma unroll
      for (int msk = 16; msk >= 1; msk >>= 1) {
        vmin = fminf(vmin, __shfl_xor(vmin, msk, 32));
        vmax = fmaxf(vmax, __shfl_xor(vmax, msk, 32));
      }
      const float gmin  = vmin * CLIPF;
      const float gmax  = vmax * CLIPF;
      const float scale = fmaxf((gmax - gmin) / MAXQF, EPSF);
#pragma unroll
      for (int j = 0; j < 4; ++j) {
        float q   = rintf(fminf(fmaxf((val[j] - gmin) / scale, 0.0f), MAXQF));
        float deq = q * scale + gmin;
        xbuf[rr[j]] = deq / mm[j];           // un-smooth + scatter (permutation)
      }
    }
    __syncthreads();

    // ---- Stage 3: async-copy token LDS -> global (coalesced) -------------
#pragma unroll
    for (int k = 0; k < 4; ++k) {
      int v    = tid + k * NTHREADS;
      int nh_i = v >> 5;
      int off  = (v & 31) * 4;
      float* gp = out + base + (size_t)nh_i * ((size_t)SL * HD) + off;
      async_store16(gp, &xbuf[v * 4]);
    }
    // Each wave reloads exactly the region it stored, so its own
    // ASYNCcnt==0 is sufficient before the next iteration reuses xbuf.
    wait_async0();
  }
}

extern "C" void kernel_launch(void* const* d_in, const int* in_sizes, int n_in,
                              void* d_out, int out_size, void* d_ws, size_t ws_size,
                              hipStream_t stream) {
  (void)in_sizes; (void)n_in; (void)out_size; (void)d_ws; (void)ws_size;
  const float* x    = (const float*)d_in[0];   // tensor  [4,32,4096,128] f32
  const float* ss   = (const float*)d_in[1];   // smooth_scale [4096] f32
  const int*   ridx = (const int*)d_in[2];     // reorder_idx  [4096] i32
  float*       out  = (float*)d_out;           // [4,32,4096,128] f32

  skvq_kernel<<<NBLOCKS, NTHREADS, 0, stream>>>(x, ss, ridx, out);
}